// WLC_loss_54546084659219
// MI455X (gfx1250) — compile-verified
//
#include <hip/hip_runtime.h>
#include <hip/hip_bf16.h>
#include <cstdint>

// ---------------------------------------------------------------------------
// Weighted-loss-of-cumulative-dilation, MI455X (gfx1250, wave32).
// Memory-bound stencil: ~1 GB traffic, ~44us at 23.3 TB/s; working set
// (~192MB) ~= global L2, so iterations 2..5 run out of L2.
// 3x3 box-sum done as banded-matrix WMMA (v_wmma_f32_16x16x4_f32 chains),
// tile staging via async global->LDS (CDNA5 path) when available.
// Scratch layout in d_ws (floats): [predA | predB | gtPart(4096) | fnPart(5*8192)]
//   -> requires ws_size >= ~128.2 MiB.
// ---------------------------------------------------------------------------

#define IMG_H 4096
#define IMG_W 4096
#define IMG_HW (IMG_H * IMG_W)

constexpr int ITERS  = 5;
constexpr int TPITCH = 132;  // 18 x 132 LDS tile (130 used cols + pad)
constexpr int VP     = 20;   // per-wave V scratch pitch (18 used + K-pad to 20)

typedef float v2f __attribute__((ext_vector_type(2)));
typedef float v8f __attribute__((ext_vector_type(8)));

// Types matching the async-LDS builtin signature:
//   (addrspace(1) v4i*, addrspace(3) v4i*, imm offset, imm cpol)
typedef int v4i __attribute__((vector_size(16)));
typedef __attribute__((address_space(1))) v4i v4i_glb;
typedef __attribute__((address_space(3))) v4i v4i_lds;

#if __has_builtin(__builtin_amdgcn_global_load_async_to_lds_b128)
#define HAVE_ASYNC_LDS 1
#else
#define HAVE_ASYNC_LDS 0
#endif

__device__ __forceinline__ void wait_async_zero() {
#if __has_builtin(__builtin_amdgcn_s_wait_asynccnt)
  __builtin_amdgcn_s_wait_asynccnt(0);
#else
  asm volatile("s_wait_asynccnt 0" ::: "memory");
#endif
}

// band(m,k) = 1 if k in {m, m+1, m+2}  (3-tap selection band)
__device__ __forceinline__ float bandv(int m, int k) {
  return (k >= m && k <= m + 2) ? 1.0f : 0.0f;
}

// ---------------------------------------------------------------------------
// One dilation+sigmoid+FN iteration. Grid: (32, 256) blocks of 256 threads.
// Each block: 16 rows x 128 cols of output; each wave: one 16x16 tile.
// ---------------------------------------------------------------------------
__global__ __launch_bounds__(256) void wlc_step_kernel(
    const float* __restrict__ predIn, float* __restrict__ predOut,
    const float* __restrict__ gt, float* __restrict__ fnPart) {
  __shared__ float tile[18 * TPITCH];      // pred tile + 1-halo (zero padded)
  __shared__ float vsAll[8 * 16 * VP];     // per-wave vertical-sum scratch
  __shared__ float red[256];

  const int tid  = threadIdx.x;
  const int bx   = blockIdx.x;             // 0..31
  const int by   = blockIdx.y;             // 0..255
  const int row0 = by * 16 - 1;            // global row of local row 0
  const int colL = bx * 128 - 1;           // global col of local col 0

  // ---- stage pred tile: edge columns (local col 0 / 129), guarded scalar ----
  for (int i = tid; i < 36; i += 256) {
    int lr = i >> 1;
    int lc = (i & 1) ? 129 : 0;
    int r = row0 + lr, c = colL + lc;
    float v = 0.0f;
    if (r >= 0 && r < IMG_H && c >= 0 && c < IMG_W)
      v = predIn[(size_t)r * IMG_W + c];
    tile[lr * TPITCH + lc] = v;
  }
  // zero-pad the center of out-of-image halo rows (top / bottom blocks)
  if (row0 < 0)
    for (int lc = 1 + tid; lc <= 128; lc += 256) tile[lc] = 0.0f;
  if (row0 + 17 >= IMG_H)
    for (int lc = 1 + tid; lc <= 128; lc += 256) tile[17 * TPITCH + lc] = 0.0f;

  // ---- center 128 cols x 18 rows: wide 16B transfers ----
#if HAVE_ASYNC_LDS
  for (int chunk = tid; chunk < 18 * 32; chunk += 256) {
    int lr = chunk >> 5, c4 = chunk & 31;
    int r = row0 + lr;
    if (r >= 0 && r < IMG_H) {
      const float* gsrc = predIn + (size_t)r * IMG_W + bx * 128 + c4 * 4;
      float* ldst = &tile[lr * TPITCH + 1 + c4 * 4];
      __builtin_amdgcn_global_load_async_to_lds_b128((v4i_glb*)gsrc,
                                                     (v4i_lds*)ldst, 0, 0);
    }
  }
  wait_async_zero();
#else
  for (int chunk = tid; chunk < 18 * 32; chunk += 256) {
    int lr = chunk >> 5, c4 = chunk & 31;
    int r = row0 + lr;
    if (r >= 0 && r < IMG_H) {
      const float4 v =
          *(const float4*)(predIn + (size_t)r * IMG_W + bx * 128 + c4 * 4);
      float* dst = &tile[lr * TPITCH + 1 + c4 * 4];
      dst[0] = v.x; dst[1] = v.y; dst[2] = v.z; dst[3] = v.w;
    }
  }
#endif
  __syncthreads();

  const int lane = tid & 31;
  const int wave = tid >> 5;
  const int n    = lane & 15;   // N coord (cols) / M coord for A-operand
  const int half = lane >> 4;   // lane half selects K pair
  const int c0   = 1 + wave * 16;  // local col of this wave's column 0
  float* vs = vsAll + wave * (16 * VP);

  // ---- vertical 3-tap as banded matmul: V = Sv(16x20) x X(20x16) ----
  v8f vacc = {0.f, 0.f, 0.f, 0.f, 0.f, 0.f, 0.f, 0.f};
#pragma unroll
  for (int kc = 0; kc < 20; kc += 4) {
    const int kb = kc + 2 * half;
    v2f a, b;
    a.x = bandv(n, kb);
    a.y = bandv(n, kb + 1);
    b.x = (kb < 18) ? tile[kb * TPITCH + c0 + n] : 0.0f;
    b.y = (kb + 1 < 18) ? tile[(kb + 1) * TPITCH + c0 + n] : 0.0f;
    vacc = __builtin_amdgcn_wmma_f32_16x16x4_f32(false, a, false, b, (short)0,
                                                 vacc, false, false);
  }

  // spill V (center cols -> slots 1..16) into per-wave LDS scratch
#pragma unroll
  for (int v = 0; v < 8; ++v) {
    int m = v + 8 * half;
    vs[m * VP + 1 + n] = vacc[v];
  }
  // halo columns (slot 0 = col-1, slot 17 = col+16) + zero K-pad slots 18/19
  {
    int m = n;
    int lc = half ? (c0 + 16) : (c0 - 1);
    float s3 = tile[m * TPITCH + lc] + tile[(m + 1) * TPITCH + lc] +
               tile[(m + 2) * TPITCH + lc];
    vs[m * VP + (half ? 17 : 0)]  = s3;
    vs[m * VP + (half ? 19 : 18)] = 0.0f;
  }

  // ---- horizontal 3-tap: Box = Vext(16x20) x Sh(20x16) ----
  v8f box = {0.f, 0.f, 0.f, 0.f, 0.f, 0.f, 0.f, 0.f};
#pragma unroll
  for (int kc = 0; kc < 20; kc += 4) {
    const int kb = kc + 2 * half;
    v2f a, b;
    a.x = vs[n * VP + kb];       // A: M = lane&15, K = kb / kb+1
    a.y = vs[n * VP + kb + 1];
    b.x = bandv(n, kb);          // Sh[k][n] = 1 iff k in {n,n+1,n+2}
    b.y = bandv(n, kb + 1);
    box = __builtin_amdgcn_wmma_f32_16x16x4_f32(false, a, false, b, (short)0,
                                                box, false, false);
  }

  // ---- clip, sharp sigmoid, FN accumulate, write pred_next ----
  float lsum = 0.0f;
  const int gr0 = by * 16;
  const int gc  = bx * 128 + wave * 16 + n;
#pragma unroll
  for (int v = 0; v < 8; ++v) {
    int m  = v + 8 * half;
    int gr = gr0 + m;
    float d  = fminf(fmaxf(box[v], 0.0f), 1.0f);
    float p  = tile[(m + 1) * TPITCH + c0 + n];
    float sg = 1.0f / (1.0f + __expf(-20.0f * (d - p - 0.5f)));
    float g  = gt[(size_t)gr * IMG_W + gc];
    float fn = g * sg;
    lsum += fn;
    predOut[(size_t)gr * IMG_W + gc] = p + fn;
  }

  // deterministic block reduction -> one partial per block
  red[tid] = lsum;
  __syncthreads();
  for (int s = 128; s > 0; s >>= 1) {
    if (tid < s) red[tid] += red[tid + s];
    __syncthreads();
  }
  if (tid == 0) fnPart[by * gridDim.x + bx] = red[0];
}

// ---------------------------------------------------------------------------
// Deterministic partial sum of a big fp32 array (for gt_sum).
// ---------------------------------------------------------------------------
__global__ __launch_bounds__(256) void wlc_sum_kernel(
    const float* __restrict__ x, float* __restrict__ part, int nElem) {
  __shared__ float red[256];
  int gid = blockIdx.x * 256 + threadIdx.x;
  int stride = gridDim.x * 256;
  float s = 0.0f;
  for (int i = gid; i < nElem; i += stride) s += x[i];
  red[threadIdx.x] = s;
  __syncthreads();
  for (int k = 128; k > 0; k >>= 1) {
    if (threadIdx.x < k) red[threadIdx.x] += red[threadIdx.x + k];
    __syncthreads();
  }
  if (threadIdx.x == 0) part[blockIdx.x] = red[0];
}

// ---------------------------------------------------------------------------
// Combine: out = sum_i (i+1)^2 * fnSum_i / gt_sum.  Single block.
// ---------------------------------------------------------------------------
__global__ __launch_bounds__(256) void wlc_final_kernel(
    const float* __restrict__ gtPart, int nGt, const float* __restrict__ fnPart,
    int nbPerIter, float* __restrict__ out) {
  __shared__ float red[256];
  const int tid = threadIdx.x;
  float s = 0.0f;
  for (int i = tid; i < nGt; i += 256) s += gtPart[i];
  red[tid] = s;
  __syncthreads();
  for (int k = 128; k > 0; k >>= 1) {
    if (tid < k) red[tid] += red[tid + k];
    __syncthreads();
  }
  const float gt_sum = red[0];
  __syncthreads();

  float acc = 0.0f;
  for (int i = tid; i < ITERS * nbPerIter; i += 256) {
    int it  = i / nbPerIter;
    float w = (float)((it + 1) * (it + 1));
    acc += w * fnPart[i];
  }
  red[tid] = acc;
  __syncthreads();
  for (int k = 128; k > 0; k >>= 1) {
    if (tid < k) red[tid] += red[tid + k];
    __syncthreads();
  }
  if (tid == 0) out[0] = red[0] / gt_sum;
}

// ---------------------------------------------------------------------------
extern "C" void kernel_launch(void* const* d_in, const int* in_sizes, int n_in,
                              void* d_out, int out_size, void* d_ws,
                              size_t ws_size, hipStream_t stream) {
  const float* pred = (const float*)d_in[0];
  const float* gt   = (const float*)d_in[1];
  float* out = (float*)d_out;

  float* ws     = (float*)d_ws;
  float* predA  = ws;
  float* predB  = ws + (size_t)IMG_HW;
  float* gtPart = ws + 2 * (size_t)IMG_HW;
  const int NGT = 4096;
  float* fnPart = gtPart + NGT;
  const int NBX = 32, NBY = 256, NB = NBX * NBY;

  wlc_sum_kernel<<<NGT, 256, 0, stream>>>(gt, gtPart, IMG_HW);

  const float* in = pred;
  float* outb = predA;
  for (int i = 0; i < ITERS; ++i) {
    wlc_step_kernel<<<dim3(NBX, NBY), 256, 0, stream>>>(
        in, outb, gt, fnPart + (size_t)i * NB);
    in = outb;
    outb = (outb == predA) ? predB : predA;
  }

  wlc_final_kernel<<<1, 256, 0, stream>>>(gtPart, NGT, fnPart, NB, out);
}